// TruePF_41455024341456
// MI455X (gfx1250) — compile-verified
//
#include <hip/hip_runtime.h>

typedef __attribute__((ext_vector_type(2))) float v2f;
typedef __attribute__((ext_vector_type(4))) float v4f;
typedef __attribute__((ext_vector_type(8))) float v8f;

#define NPART 131072          // N (2^17)
#define BROWS 128             // B
#define ELEMS_PER_BLOCK 1024  // 256 threads * 4 floats
#define BLOCKS_PER_ROW (NPART / ELEMS_PER_BLOCK)  // 128
#define TOTAL_BLOCKS (BROWS * BLOCKS_PER_ROW)     // 16384

// Pass 1: transition + unnormalized weights + per-block partial sums.
__global__ __launch_bounds__(256) void pf_main(
    const float* __restrict__ particles,
    const float* __restrict__ weights,
    const float* __restrict__ observation,
    const float* __restrict__ noise,
    const int* __restrict__ tstep,
    float* __restrict__ out_np,
    float* __restrict__ out_w,
    float* __restrict__ partials) {
  const int tid = threadIdx.x;
  const int row = (int)blockIdx.x / BLOCKS_PER_ROW;
  const int chunk = (int)blockIdx.x % BLOCKS_PER_ROW;
  const int base = row * NPART + chunk * ELEMS_PER_BLOCK + tid * 4;

  // Streamed inputs: non-temporal (no reuse; keep them out of L2).
  const v4f p  = __builtin_nontemporal_load((const v4f*)(particles + base));
  const v4f pw = __builtin_nontemporal_load((const v4f*)(weights + base));
  const v4f nz = __builtin_nontemporal_load((const v4f*)(noise + base));
  const float obs = observation[row];
  const float drift = 8.0f * __cosf(1.2f * (float)tstep[0]);
  const float sqrt10 = 3.1622776601683795f;
  const float neg_half_log2pi = -0.91893853320467274f;

  v4f np4, w4;
  float local = 0.0f;
#pragma unroll
  for (int j = 0; j < 4; ++j) {
    float x = p[j];
    float mean = 0.5f * x + 25.0f * (x / (x * x + 1.0f)) + drift;
    float xn = mean + nz[j] * sqrt10;          // new particle
    float om = xn * xn * 0.05f;                // x'^2 / 20
    float d = obs - om;
    float lp = -0.5f * d * d + neg_half_log2pi;
    float w = pw[j] * __expf(lp);
    np4[j] = xn;
    w4[j] = w;
    local += w;
  }

  // new_particles: streamed out, non-temporal.
  __builtin_nontemporal_store(np4, (v4f*)(out_np + base));
  // w: regular-temporal so the 64MB array stays in the 192MB L2 for pass 3.
  *(v4f*)(out_w + base) = w4;

  // Exact f32 wave32 reduction via V_WMMA_F32_16X16X4_F32:
  // A[m][0]=local(lane m), A[m][2]=local(lane m+16), A[m][1]=A[m][3]=0;
  // B = ones(4x16)  =>  D[m][n] = local(m) + local(m+16).
  v2f a; a.x = local; a.y = 0.0f;
  v2f bones; bones.x = 1.0f; bones.y = 1.0f;
  v8f c0 = {};
  v8f dm = __builtin_amdgcn_wmma_f32_16x16x4_f32(
      false, a, false, bones, (short)0, c0, false, false);
  // Lanes 0-15 hold D rows 0..7 (partials 0-7 + 16-23), lanes 16-31 rows 8..15.
  float s = dm[0] + dm[1] + dm[2] + dm[3] + dm[4] + dm[5] + dm[6] + dm[7];
  s += __shfl_xor(s, 16, 32);  // combine half-wave sums -> full wave sum

  __shared__ float wsum[8];
  const int lane = tid & 31;
  const int wave = tid >> 5;
  if (lane == 0) wsum[wave] = s;
  __syncthreads();
  if (tid == 0) {
    float t = 0.0f;
#pragma unroll
    for (int j = 0; j < 8; ++j) t += wsum[j];
    partials[blockIdx.x] = t;  // index == row*BLOCKS_PER_ROW + chunk
  }
}

// Pass 2: deterministic fixed-tree reduction of 128 partials per row.
__global__ __launch_bounds__(128) void pf_reduce(
    const float* __restrict__ partials, float* __restrict__ rowsum) {
  const int row = blockIdx.x;
  const int tid = threadIdx.x;
  __shared__ float sm[128];
  sm[tid] = partials[row * BLOCKS_PER_ROW + tid];
  __syncthreads();
#pragma unroll
  for (int off = 64; off > 0; off >>= 1) {
    if (tid < off) sm[tid] += sm[tid + off];
    __syncthreads();
  }
  if (tid == 0) rowsum[row] = sm[0];
}

// Pass 3: normalize w (reads should hit L2; final store non-temporal).
__global__ __launch_bounds__(256) void pf_norm(
    float* __restrict__ out_w, const float* __restrict__ rowsum) {
  const int base = ((int)blockIdx.x * 256 + (int)threadIdx.x) * 4;
  const int row = base >> 17;  // N = 2^17
  const float inv = 1.0f / rowsum[row];
  v4f w = *(const v4f*)(out_w + base);
  w *= inv;
  __builtin_nontemporal_store(w, (v4f*)(out_w + base));
}

extern "C" void kernel_launch(void* const* d_in, const int* in_sizes, int n_in,
                              void* d_out, int out_size, void* d_ws, size_t ws_size,
                              hipStream_t stream) {
  const float* particles   = (const float*)d_in[0];
  const float* weights     = (const float*)d_in[1];
  const float* observation = (const float*)d_in[2];
  const float* noise       = (const float*)d_in[3];
  // d_in[4] (uniforms) is dead in the reference: resample output never reaches outputs.
  const int* tstep         = (const int*)d_in[5];

  float* out_np = (float*)d_out;                                   // [B*N]
  float* out_w  = (float*)d_out + (size_t)BROWS * NPART;           // [B*N]

  float* partials = (float*)d_ws;                 // TOTAL_BLOCKS floats
  float* rowsum   = (float*)d_ws + TOTAL_BLOCKS;  // BROWS floats

  pf_main<<<TOTAL_BLOCKS, 256, 0, stream>>>(particles, weights, observation,
                                            noise, tstep, out_np, out_w,
                                            partials);
  pf_reduce<<<BROWS, 128, 0, stream>>>(partials, rowsum);
  pf_norm<<<(BROWS * NPART) / ELEMS_PER_BLOCK, 256, 0, stream>>>(out_w, rowsum);
}